// CRFLoss_8323646620218
// MI455X (gfx1250) — compile-verified
//
#include <hip/hip_runtime.h>
#include <math.h>

#define Bn 16
#define Vn 8
#define Sn 256
#define Kn 66
#define KP 68   // K padded to multiple of 4 (17 K-tiles of 4)
#define NP 80   // N padded to 5 tiles of 16
#define NSEQ (Bn * Vn)  // 128 sequences

typedef __attribute__((ext_vector_type(2))) float v2f;
typedef __attribute__((ext_vector_type(8))) float v8f;

// ---------------------------------------------------------------------------
// CRF forward (log-partition) via f32 WMMA in exp-space.
// One wave (32 lanes) per workgroup handles 16 sequences; 8 workgroups total.
// alpha recursion: s[m,j] = sum_i exp(alpha[m,i]-mx[m]) * exp(trans[i,j])
//                  alpha'[m,j] = mx[m] + log(s[m,j]) + logit[m,t,j]
// ---------------------------------------------------------------------------
__global__ __launch_bounds__(32)
void crf_forward(const float* __restrict__ score,
                 const int*   __restrict__ v_label,
                 const int*   __restrict__ orig_l,
                 const float* __restrict__ start_t,
                 const float* __restrict__ trans,
                 const float* __restrict__ end_t,
                 float*       __restrict__ logz) {
  __shared__ float sExpT[KP][NP];   // exp(trans), zero-padded
  __shared__ float sAlpha[16][KP];  // alpha per sequence
  __shared__ float sEa[16][KP];     // exp(alpha - mx), zero-padded (A matrix)
  __shared__ float sS[16][NP];      // WMMA result D

  const int lane = threadIdx.x;   // 0..31
  const int g    = blockIdx.x;    // 0..7

  // Precompute exp(trans) with zero padding.
  for (int idx = lane; idx < KP * NP; idx += 32) {
    int i = idx / NP, j = idx - i * NP;
    float v = 0.0f;
    if (i < Kn && j < Kn) v = expf(trans[i * Kn + j]);
    sExpT[i][j] = v;
  }

  // Per-sequence state kept in lane-private registers of lanes 0..15.
  const float* base = nullptr;
  int L = 0;
  if (lane < 16) {
    int n = g * 16 + lane;
    int b = n / Vn, v = n - b * Vn;
    int s1 = v_label[b * Vn + v];
    base = score + ((size_t)(b * Sn + s1)) * Sn * Kn;  // a_score[m, :, :]
    L = orig_l[b];
    // alpha0 = start_t + logits[:,0]
    for (int j = 0; j < KP; ++j)
      sAlpha[lane][j] = (j < Kn) ? (start_t[j] + base[j]) : -1e30f;
  }
  __syncthreads();

  const int arow = lane & 15;
  const int ksel = (lane >> 4) << 1;  // 0 for lanes 0-15, 2 for lanes 16-31

  float mx = 0.0f;
  for (int t = 1; t < Sn; ++t) {
    // per-sequence max + exp into the A-matrix staging buffer
    if (lane < 16) {
      float m_ = -1e30f;
      for (int j = 0; j < Kn; ++j) m_ = fmaxf(m_, sAlpha[lane][j]);
      mx = m_;
      for (int j = 0; j < KP; ++j)
        sEa[lane][j] = (j < Kn) ? expf(sAlpha[lane][j] - m_) : 0.0f;
    }
    __syncthreads();

    // Load the 17 A-fragments (16x4 f32 layout: lanes 0-15 -> K=0,1; 16-31 -> K=2,3)
    v2f afrag[17];
#pragma unroll
    for (int k = 0; k < 17; ++k) {
      int kb = 4 * k + ksel;
      afrag[k].x = sEa[arow][kb];
      afrag[k].y = sEa[arow][kb + 1];
    }

    // D(16x80) = A(16x68) * B(68x80), 5 N-tiles x 17 K-tiles
    for (int nt = 0; nt < 5; ++nt) {
      v8f acc = {0.0f, 0.0f, 0.0f, 0.0f, 0.0f, 0.0f, 0.0f, 0.0f};
      int ncol = nt * 16 + arow;
#pragma unroll
      for (int k = 0; k < 17; ++k) {
        int kb = 4 * k + ksel;
        v2f bf;
        bf.x = sExpT[kb][ncol];
        bf.y = sExpT[kb + 1][ncol];
        acc = __builtin_amdgcn_wmma_f32_16x16x4_f32(
            /*neg_a=*/false, afrag[k], /*neg_b=*/false, bf,
            /*c_mod=*/(short)0, acc, /*reuse_a=*/false, /*reuse_b=*/false);
      }
      // Write D: VGPR r -> row r (lanes 0-15) / row r+8 (lanes 16-31)
      int drow = (lane < 16) ? 0 : 8;
#pragma unroll
      for (int r = 0; r < 8; ++r) sS[drow + r][ncol] = acc[r];
    }
    __syncthreads();

    // Masked alpha update: alpha = mask ? mx + log(s) + logit_t : alpha
    if (lane < 16 && t < L) {
      const float* lg = base + (size_t)t * Kn;
      for (int j = 0; j < Kn; ++j)
        sAlpha[lane][j] = mx + logf(sS[lane][j]) + lg[j];
    }
    __syncthreads();
  }

  // log_z = logsumexp(alpha + end_t)
  if (lane < 16) {
    float m_ = -1e30f;
    for (int j = 0; j < Kn; ++j) m_ = fmaxf(m_, sAlpha[lane][j] + end_t[j]);
    float s = 0.0f;
    for (int j = 0; j < Kn; ++j) s += expf(sAlpha[lane][j] + end_t[j] - m_);
    logz[g * 16 + lane] = m_ + logf(s);
  }
}

// ---------------------------------------------------------------------------
// Gold-path score per sequence (128 threads).
// ---------------------------------------------------------------------------
__global__ __launch_bounds__(128)
void gold_score(const float* __restrict__ score,
                const int*   __restrict__ v_label,
                const int*   __restrict__ orig_l,
                const int*   __restrict__ role_label,
                const float* __restrict__ start_t,
                const float* __restrict__ trans,
                const float* __restrict__ end_t,
                float*       __restrict__ out) {
  int n = threadIdx.x;
  if (n >= NSEQ) return;
  int b = n / Vn, v = n - b * Vn;
  int s1 = v_label[b * Vn + v];
  const float* lg = score + ((size_t)(b * Sn + s1)) * Sn * Kn;
  const int* tg = role_label + (size_t)n * Sn;
  int L = orig_l[b];

  float sc = start_t[tg[0]];
  for (int t = 0; t < Sn - 1; ++t) {
    int tagt = tg[t];
    if (t < L)     sc += lg[(size_t)t * Kn + tagt];        // emit * mask
    if (t + 1 < L) sc += trans[tagt * Kn + tg[t + 1]];     // trans * mask
  }
  sc += end_t[tg[L - 1]];                                  // end at last valid
  if (Sn - 1 < L) sc += lg[(size_t)(Sn - 1) * Kn + tg[Sn - 1]];
  out[n] = sc;
}

// ---------------------------------------------------------------------------
// loss = sum_n (gold[n] - logz[n]) / 128
// ---------------------------------------------------------------------------
__global__ __launch_bounds__(128)
void reduce_loss(const float* __restrict__ gold,
                 const float* __restrict__ logz,
                 float* __restrict__ out) {
  __shared__ float buf[128];
  int t = threadIdx.x;
  buf[t] = gold[t] - logz[t];
  __syncthreads();
  for (int s = 64; s > 0; s >>= 1) {
    if (t < s) buf[t] += buf[t + s];
    __syncthreads();
  }
  if (t == 0) out[0] = buf[0] / (float)NSEQ;
}

// ---------------------------------------------------------------------------
// pred_idx = argmax(log_pa, axis=-1); one thread per (b,s1,s2).
// Rows are 66 f32 = 264 B (8-byte aligned) -> float2 loads, strict-> compare
// preserves first-max semantics.
// ---------------------------------------------------------------------------
__global__ __launch_bounds__(256)
void argmax_k(const float* __restrict__ lp, float* __restrict__ out, int total) {
  int p = blockIdx.x * blockDim.x + threadIdx.x;
  if (p >= total) return;
  const v2f* v2 = (const v2f*)(lp + (size_t)p * Kn);
  float best = -1e38f;
  int bi = 0;
#pragma unroll
  for (int j2 = 0; j2 < Kn / 2; ++j2) {
    v2f x = v2[j2];
    if (x.x > best) { best = x.x; bi = 2 * j2; }
    if (x.y > best) { best = x.y; bi = 2 * j2 + 1; }
  }
  out[p] = (float)bi;
}

extern "C" void kernel_launch(void* const* d_in, const int* in_sizes, int n_in,
                              void* d_out, int out_size, void* d_ws, size_t ws_size,
                              hipStream_t stream) {
  const float* log_pa     = (const float*)d_in[0];
  const float* score      = (const float*)d_in[1];
  const int*   v_label    = (const int*)d_in[2];
  /* d_in[3] = v_l (always V) unused */
  const int*   orig_l     = (const int*)d_in[4];
  const int*   role_label = (const int*)d_in[5];
  const float* start_t    = (const float*)d_in[6];
  const float* trans      = (const float*)d_in[7];
  const float* end_t      = (const float*)d_in[8];

  float* out = (float*)d_out;
  float* ws  = (float*)d_ws;
  float* wsGold = ws;           // [0,128)
  float* wsLogz = ws + NSEQ;    // [128,256)

  gold_score<<<1, 128, 0, stream>>>(score, v_label, orig_l, role_label,
                                    start_t, trans, end_t, wsGold);
  crf_forward<<<8, 32, 0, stream>>>(score, v_label, orig_l,
                                    start_t, trans, end_t, wsLogz);
  reduce_loss<<<1, 128, 0, stream>>>(wsGold, wsLogz, out);

  int total = Bn * Sn * Sn;
  argmax_k<<<(total + 255) / 256, 256, 0, stream>>>(log_pa, out + 1, total);
}